// JBLD_DisNet_9423158247639
// MI455X (gfx1250) — compile-verified
//
#include <hip/hip_runtime.h>
#include <math.h>

// JBLD distance for batched 32x32 SPD pairs on gfx1250 (MI455X), fully
// register-resident blocked Cholesky.
//
// dist[n] = logdet((A+B)/2) - 0.5*logdet(A) - 0.5*logdet(B)
// logdet  = sum_k log(d_k)   (d_k = Cholesky pivots before sqrt)
//
// One wave32 per sample, no LDS:
//  * panel 1 (cols 0..15, rows 0..31): lane owns its row in 16 VGPRs;
//    pivots/column broadcasts via v_readlane (uniform, no memory).
//  * Schur update A22 -= L21*L21^T: 4 chained V_WMMA_F32_16X16X4_F32 (pure
//    fp32 matrix pipe, no precision loss). A/B operands gathered from the
//    upper-half lanes' registers with ds_bpermute shuffles; C operand loaded
//    from global directly in the WMMA C layout.
//  * A22' is symmetric, so the column-major WMMA output redistributes to
//    row-major panel-2 registers with 8 shfl_xor(16) + selects.
//  * panel 2 (16x16 trailing block) register-resident, duplicated across
//    half-waves so EXEC stays full throughout (WMMA requirement).
// Memory: 6KB/sample (left 32x16 panel + A22; A12 skipped by symmetry).

typedef float v2f __attribute__((ext_vector_type(2)));
typedef float v8f __attribute__((ext_vector_type(8)));

#define SAMPLES_PER_BLOCK 8   // 8 waves * 32 lanes = 256 threads

// Uniform broadcast from a (compile-time constant) lane: v_readlane -> SGPR.
__device__ __forceinline__ float bcast(float v, int srclane) {
  return __int_as_float(__builtin_amdgcn_readlane(__float_as_int(v), srclane));
}

struct HalfMat {
  float r[16];  // row `lane` of M[:, 0:16]  (panel-1 operand, rows 0..31)
  v8f   c;      // M[16:32, 16:32] in WMMA 16x16 f32 C layout
};

__device__ __forceinline__ HalfMat load_half(const float* __restrict__ g, int lane) {
  HalfMat h;
  const int ln = lane & 15;
  const int hi = lane >> 4;
  const float4* rowp = reinterpret_cast<const float4*>(g + lane * 32);
#pragma unroll
  for (int q = 0; q < 4; ++q) {           // global_load_b128 x4 (cols 0..15)
    float4 v = rowp[q];
    h.r[4 * q + 0] = v.x; h.r[4 * q + 1] = v.y;
    h.r[4 * q + 2] = v.z; h.r[4 * q + 3] = v.w;
  }
#pragma unroll
  for (int v = 0; v < 8; ++v)             // A22 in WMMA C layout: m=v+8*hi, n=ln
    h.c[v] = g[(16 + v + 8 * hi) * 32 + 16 + ln];
  return h;
}

// In-register blocked Cholesky; returns t = sum_k log(d_k) over all 32 pivots
// (wave-uniform). sumlog(diag L) = 0.5 * t, logdet = t.
__device__ __forceinline__ float chol32_logpivots(HalfMat h, int lane) {
  const int ln = lane & 15;
  const int hi = lane >> 4;
  float t = 0.0f;
  float prod = 1.0f;  // product of <=4 pivots; d_k >= lambda_min >= 32, no overflow

  // ---- panel 1: columns 0..15, rows = lane (0..31), fully unrolled ----
#pragma unroll
  for (int k = 0; k < 16; ++k) {
    float dk = bcast(h.r[k], k);                       // pivot (uniform)
    prod *= dk;
    if ((k & 3) == 3) { t += logf(prod); prod = 1.0f; }
    float inv = rsqrtf(dk);
    h.r[k] = (lane == k) ? dk * inv : h.r[k] * inv;    // sqrt / scale column
#pragma unroll
    for (int j = k + 1; j < 16; ++j) {
      float ljk = bcast(h.r[k], j);                    // L[j][k], lane j > k: valid
      h.r[j] = fmaf(-h.r[k], ljk, h.r[j]);             // lanes < k: garbage, never read
    }
  }

  // ---- WMMA Schur complement: c -= L21 * L21^T  (K=16 as 4 chunks) ----
  // A operand: lane<16 -> M=lane K={4ch,4ch+1}; lane>=16 -> M=lane-16 K={+2,+3}.
  // L21 row m lives in lane 16+m's registers -> gather via bpermute from the
  // mirrored upper-half lane. B (= L21^T chunk) has identical register values.
  const int src = 16 + ln;
#pragma unroll
  for (int ch = 0; ch < 4; ++ch) {
    float t0 = __shfl(h.r[4 * ch + 0], src, 32);
    float t1 = __shfl(h.r[4 * ch + 1], src, 32);
    float t2 = __shfl(h.r[4 * ch + 2], src, 32);
    float t3 = __shfl(h.r[4 * ch + 3], src, 32);
    v2f b, a;
    b.x = hi ? t2 : t0;
    b.y = hi ? t3 : t1;
    a.x = -b.x;                                        // fp32 WMMA: only CNeg exists,
    a.y = -b.y;                                        // so negate A in VALU
    // 8 args: (neg_a, A, neg_b, B, c_mod, C, reuse_a, reuse_b)
    h.c = __builtin_amdgcn_wmma_f32_16x16x4_f32(false, a, false, b,
                                                (short)0, h.c, false, false);
  }

  // ---- symmetric redistribute: row2[j] = S22'[ln][j], duplicated halves ----
  // WMMA D: lane holds column n=ln, rows m=v+8*hi. S22' symmetric => column
  // ln == row ln; swap halves with shfl_xor(16) to assemble the full row.
  float row2[16];
#pragma unroll
  for (int v = 0; v < 8; ++v) {
    float q = __shfl_xor(h.c[v], 16, 32);
    row2[v]     = hi ? q      : h.c[v];
    row2[8 + v] = hi ? h.c[v] : q;
  }

  // ---- panel 2: trailing 16x16 block, rows duplicated across half-waves ----
#pragma unroll
  for (int k = 0; k < 16; ++k) {
    float dk = bcast(row2[k], k);
    prod *= dk;
    if ((k & 3) == 3) { t += logf(prod); prod = 1.0f; }
    float inv = rsqrtf(dk);
    row2[k] = (ln == k) ? dk * inv : row2[k] * inv;
#pragma unroll
    for (int j = k + 1; j < 16; ++j) {
      float ljk = bcast(row2[k], j);
      row2[j] = fmaf(-row2[k], ljk, row2[j]);
    }
  }
  return t;
}

__global__ __launch_bounds__(SAMPLES_PER_BLOCK * 32)
void jbld_kernel(const float* __restrict__ x, float* __restrict__ out, int nsamp) {
  const int lane = threadIdx.x & 31;
  const int wave = threadIdx.x >> 5;
  const int n    = blockIdx.x * SAMPLES_PER_BLOCK + wave;   // wave-uniform
  if (n >= nsamp) return;

  const float* base = x + (size_t)n * 2048;

  // Load both halves up-front: B's loads overlap A's factorization, and the
  // kept-live registers let us form (A+B)/2 without re-reading global.
  HalfMat ha = load_half(base, lane);
  HalfMat hb = load_half(base + 1024, lane);

  float tA = chol32_logpivots(ha, lane);   // by-value copy; originals kept
  float tB = chol32_logpivots(hb, lane);

  HalfMat hc;
#pragma unroll
  for (int j = 0; j < 16; ++j) hc.r[j] = 0.5f * (ha.r[j] + hb.r[j]);
  hc.c = 0.5f * (ha.c + hb.c);
  float tC = chol32_logpivots(hc, lane);

  // logdet = t;  dist = tC - 0.5*(tA + tB). t is wave-uniform.
  if (lane == 0) out[n] = tC - 0.5f * (tA + tB);
}

extern "C" void kernel_launch(void* const* d_in, const int* in_sizes, int n_in,
                              void* d_out, int out_size, void* d_ws, size_t ws_size,
                              hipStream_t stream) {
  const float* x   = (const float*)d_in[0];
  float* out       = (float*)d_out;
  const int nsamp  = in_sizes[0] / 2048;                    // N = 65536
  const int blocks = (nsamp + SAMPLES_PER_BLOCK - 1) / SAMPLES_PER_BLOCK;
  jbld_kernel<<<blocks, SAMPLES_PER_BLOCK * 32, 0, stream>>>(x, out, nsamp);
}